// VGAEModel_10024453669132
// MI455X (gfx1250) — compile-verified
//
#include <hip/hip_runtime.h>
#include <math.h>

typedef __attribute__((ext_vector_type(2))) float v2f;
typedef __attribute__((ext_vector_type(8))) float v8f;

// ---------------------------------------------------------------- utilities
__global__ void zero_f32(float* __restrict__ p, int n) {
    int i = blockIdx.x * blockDim.x + threadIdx.x;
    if (i < n) p[i] = 0.0f;
}

__global__ void count_deg(const int* __restrict__ src, const int* __restrict__ dst,
                          float* __restrict__ degO, float* __restrict__ degI, int E) {
    int e = blockIdx.x * blockDim.x + threadIdx.x;
    if (e < E) {
        atomicAdd(&degO[src[e]], 1.0f);   // global_atomic_add_f32, no return
        atomicAdd(&degI[dst[e]], 1.0f);
    }
}

__global__ void deg_to_rsqrt(float* __restrict__ p, int n) {
    int i = blockIdx.x * blockDim.x + threadIdx.x;
    if (i < n) p[i] = rsqrtf(fmaxf(p[i], 1.0f));
}

// ------------------------------------------------ fp32 WMMA GEMM  C = A @ W
// A: [M,K] row-major, W: [K,N] row-major, C: [M,N]. One 16x16 tile per wave.
// M%16==0, K%4==0, N%16==0 guaranteed by the problem sizes (10000 = 625*16).
__global__ __launch_bounds__(32) void gemm_f32_wmma(
    const float* __restrict__ A, const float* __restrict__ W,
    float* __restrict__ C, int K, int N)
{
    const int lane = threadIdx.x;      // 0..31, wave32
    const int half = lane >> 4;        // 0|1 : K sub-pair select
    const int l16  = lane & 15;
    const int m0 = blockIdx.y * 16;
    const int n0 = blockIdx.x * 16;

    v8f acc = {};
    const float* Arow = A + (size_t)(m0 + l16) * K;
    for (int k0 = 0; k0 < K; k0 += 4) {
        const int ka = k0 + half * 2;
        v2f a, b;
        a.x = Arow[ka];
        a.y = Arow[ka + 1];
        b.x = W[(size_t)ka * N + n0 + l16];
        b.y = W[(size_t)(ka + 1) * N + n0 + l16];
        acc = __builtin_amdgcn_wmma_f32_16x16x4_f32(
                  false, a, false, b, (short)0, acc, false, false);
    }
#pragma unroll
    for (int g = 0; g < 8; ++g)
        C[(size_t)(m0 + g + 8 * half) * N + n0 + l16] = acc[g];
}

// ------------------------------------------- edge gather + scaled scatter-add
// agg[dst[e], f] += x[src[e], f] * sOut[src[e]]   (one edge per block)
__global__ void scatter_scaled(const float* __restrict__ x, const int* __restrict__ src,
                               const int* __restrict__ dst, const float* __restrict__ sOut,
                               float* __restrict__ agg, int F) {
    const int e = blockIdx.x;
    const int s = src[e], d = dst[e];
    const float sc = sOut[s];
    const int f = threadIdx.x;
    atomicAdd(&agg[(size_t)d * F + f], x[(size_t)s * F + f] * sc);
}

// h = relu(agg * rsqrt_in[row] + b)   (in place)
__global__ void ep_relu(float* __restrict__ agg, const float* __restrict__ rin,
                        const float* __restrict__ b, int F) {
    const int i = blockIdx.x;
    const int f = threadIdx.x;
    const size_t idx = (size_t)i * F + f;
    const float v = agg[idx] * rin[i] + b[f];
    agg[idx] = v > 0.0f ? v : 0.0f;
}

// z = (aggM*rin + bM) + noise * exp(aggS*rin + bS)
__global__ void make_z(const float* __restrict__ aggM, const float* __restrict__ aggS,
                       const float* __restrict__ rin, const float* __restrict__ bM,
                       const float* __restrict__ bS, const float* __restrict__ noise,
                       float* __restrict__ z, int F) {
    const int i = blockIdx.x;
    const int f = threadIdx.x;
    const size_t idx = (size_t)i * F + f;
    const float r = rin[i];
    const float mean = aggM[idx] * r + bM[f];
    const float lstd = aggS[idx] * r + bS[f];
    z[idx] = mean + noise[idx] * expf(lstd);
}

// ------------------------------------ decoder: out = sigmoid(Z @ Z^T), fp32 WMMA
// One wave computes a 16x64 strip (4 accumulators); A-fragment reused 4x.
// For Z@Z^T the A(16x4) and B(4x16) fragment loads share identical index math.
// Output (400 MB, written once, never re-read) uses non-temporal stores so the
// stream doesn't contend with the L2-resident z (5 MB in 192 MB L2).
__global__ __launch_bounds__(32) void decoder_wmma(
    const float* __restrict__ Z, float* __restrict__ out, int Nn, int Kd)
{
    const int lane = threadIdx.x;
    const int half = lane >> 4;
    const int l16  = lane & 15;
    const int m0     = blockIdx.y * 16;      // exact: Nn % 16 == 0
    const int ntile0 = blockIdx.x * 4;       // group of 4 N-tiles
    const int ntiles = Nn >> 4;

    v8f acc[4] = {{}, {}, {}, {}};
    for (int k0 = 0; k0 < Kd; k0 += 4) {
        const int ka = k0 + half * 2;
        v2f a;
        a.x = Z[(size_t)(m0 + l16) * Kd + ka];
        a.y = Z[(size_t)(m0 + l16) * Kd + ka + 1];
#pragma unroll
        for (int t = 0; t < 4; ++t) {
            const int nt = ntile0 + t;
            if (nt < ntiles) {               // block-uniform: no divergence
                const int n0 = nt * 16;
                v2f b;
                b.x = Z[(size_t)(n0 + l16) * Kd + ka];
                b.y = Z[(size_t)(n0 + l16) * Kd + ka + 1];
                acc[t] = __builtin_amdgcn_wmma_f32_16x16x4_f32(
                             false, a, false, b, (short)0, acc[t], false, false);
            }
        }
    }
#pragma unroll
    for (int t = 0; t < 4; ++t) {
        const int nt = ntile0 + t;
        if (nt >= ntiles) continue;
        const int n0 = nt * 16;
#pragma unroll
        for (int g = 0; g < 8; ++g) {
            const int row = m0 + g + 8 * half;
            const float v = acc[t][g];
            const float s = 1.0f / (1.0f + expf(-v));
            __builtin_nontemporal_store(s, &out[(size_t)row * Nn + n0 + l16]);
        }
    }
}

// --------------------------------------------------------------------- driver
extern "C" void kernel_launch(void* const* d_in, const int* in_sizes, int n_in,
                              void* d_out, int out_size, void* d_ws, size_t ws_size,
                              hipStream_t stream) {
    const float* feat  = (const float*)d_in[0];
    const int*   src   = (const int*)  d_in[1];
    const int*   dst   = (const int*)  d_in[2];
    const float* noise = (const float*)d_in[3];
    const float* W1    = (const float*)d_in[4];
    const float* b1    = (const float*)d_in[5];
    const float* W2    = (const float*)d_in[6];
    const float* b2    = (const float*)d_in[7];
    const float* W3    = (const float*)d_in[8];
    const float* b3    = (const float*)d_in[9];
    float* out = (float*)d_out;

    const int E  = in_sizes[1];              // 320000
    const int H1 = in_sizes[5];              // 256
    const int H2 = in_sizes[7];              // 128
    const int N  = in_sizes[3] / H2;         // 10000
    const int IN = in_sizes[0] / N;          // 512

    // workspace layout (floats)
    float* w = (float*)d_ws;
    float* rOut = w;                 w += N;              // rsqrt(deg_out)
    float* rIn  = w;                 w += N;              // rsqrt(deg_in)
    float* XW1  = w;                 w += (size_t)N * H1; // X @ W1
    float* h    = w;                 w += (size_t)N * H1; // agg1 -> relu h
    float* hW2  = w;                 w += (size_t)N * H2;
    float* hW3  = w;                 w += (size_t)N * H2;
    float* agg2 = w;                 w += (size_t)N * H2;
    float* agg3 = w;                 w += (size_t)N * H2; // contiguous with agg2
    float* z    = w;                 w += (size_t)N * H2;

    // 1) degrees -> rsqrt(clamp(deg,1))
    zero_f32<<<(2 * N + 255) / 256, 256, 0, stream>>>(rOut, 2 * N); // rOut+rIn contiguous
    count_deg<<<(E + 255) / 256, 256, 0, stream>>>(src, dst, rOut, rIn, E);
    deg_to_rsqrt<<<(2 * N + 255) / 256, 256, 0, stream>>>(rOut, 2 * N);

    // 2) layer 1: XW1 = X @ W1 (apply W before aggregation; GraphConv is linear)
    gemm_f32_wmma<<<dim3(H1 / 16, N / 16), 32, 0, stream>>>(feat, W1, XW1, IN, H1);
    zero_f32<<<((int)((size_t)N * H1) + 255) / 256, 256, 0, stream>>>(h, N * H1);
    scatter_scaled<<<E, H1, 0, stream>>>(XW1, src, dst, rOut, h, H1);
    ep_relu<<<N, H1, 0, stream>>>(h, rIn, b1, H1);

    // 3) layers 2/3: mean and log_std branches
    gemm_f32_wmma<<<dim3(H2 / 16, N / 16), 32, 0, stream>>>(h, W2, hW2, H1, H2);
    gemm_f32_wmma<<<dim3(H2 / 16, N / 16), 32, 0, stream>>>(h, W3, hW3, H1, H2);
    zero_f32<<<((int)(2 * (size_t)N * H2) + 255) / 256, 256, 0, stream>>>(agg2, 2 * N * H2);
    scatter_scaled<<<E, H2, 0, stream>>>(hW2, src, dst, rOut, agg2, H2);
    scatter_scaled<<<E, H2, 0, stream>>>(hW3, src, dst, rOut, agg3, H2);

    // 4) reparameterization
    make_z<<<N, H2, 0, stream>>>(agg2, agg3, rIn, b2, b3, noise, z, H2);

    // 5) decoder: sigmoid(z z^T), store-bound (400 MB); z is L2-resident
    const int mtiles = N / 16;                       // 625
    const int ngroups = (mtiles + 3) / 4;            // 157
    decoder_wmma<<<dim3(ngroups, mtiles), 32, 0, stream>>>(z, out, N, H2);
}